// DenoisingDiffusion_17025250361520
// MI455X (gfx1250) — compile-verified
//
#include <hip/hip_runtime.h>
#include <hip/hip_bf16.h>
#include <stdint.h>

typedef __attribute__((ext_vector_type(2))) float v2f;
typedef __attribute__((ext_vector_type(8))) float v8f;

// ---------------------------------------------------------------------------
// Threefry2x32 (JAX classic path): 20 rounds, rotations [13,15,26,6]/[17,29,16,24]
// ---------------------------------------------------------------------------
__device__ __forceinline__ uint32_t rotl32(uint32_t x, int n) {
    return (x << n) | (x >> (32 - n));
}

__device__ __forceinline__ void tf_round4(uint32_t& x0, uint32_t& x1,
                                          int r0, int r1, int r2, int r3) {
    x0 += x1; x1 = rotl32(x1, r0); x1 ^= x0;
    x0 += x1; x1 = rotl32(x1, r1); x1 ^= x0;
    x0 += x1; x1 = rotl32(x1, r2); x1 ^= x0;
    x0 += x1; x1 = rotl32(x1, r3); x1 ^= x0;
}

__device__ __forceinline__ void threefry2x32(uint32_t k0, uint32_t k1,
                                             uint32_t c0, uint32_t c1,
                                             uint32_t& o0, uint32_t& o1) {
    uint32_t ks0 = k0, ks1 = k1, ks2 = k0 ^ k1 ^ 0x1BD11BDAu;
    uint32_t x0 = c0 + ks0, x1 = c1 + ks1;
    tf_round4(x0, x1, 13, 15, 26, 6);  x0 += ks1; x1 += ks2 + 1u;
    tf_round4(x0, x1, 17, 29, 16, 24); x0 += ks2; x1 += ks0 + 2u;
    tf_round4(x0, x1, 13, 15, 26, 6);  x0 += ks0; x1 += ks1 + 3u;
    tf_round4(x0, x1, 17, 29, 16, 24); x0 += ks1; x1 += ks2 + 4u;
    tf_round4(x0, x1, 13, 15, 26, 6);  x0 += ks2; x1 += ks0 + 5u;
    o0 = x0; o1 = x1;
}

__device__ __forceinline__ float block_reduce_sum_256(float v) {
    __shared__ float s[256];
    s[threadIdx.x] = v;
    __syncthreads();
    for (int o = 128; o > 0; o >>= 1) {
        if ((int)threadIdx.x < o) s[threadIdx.x] += s[threadIdx.x + o];
        __syncthreads();
    }
    return s[0];
}

// ---------------------------------------------------------------------------
// Stage 0: subkeys for jax.random.split(key(42), t+1)
// split counts = iota(2*num), halves layout: bits[i]=o0(i, i+num), bits[i+num]=o1
// ---------------------------------------------------------------------------
__global__ void subkey_kernel(const int* __restrict__ tptr, uint32_t* __restrict__ bits) {
    int num = *tptr + 1;                // <= 100
    int i = threadIdx.x;
    if (i < num) {
        uint32_t o0, o1;
        threefry2x32(0u, 42u, (uint32_t)i, (uint32_t)(i + num), o0, o1);
        bits[i] = o0;
        bits[i + num] = o1;
    }
}

// ---------------------------------------------------------------------------
// Stage 1: q_sample — 51 sequential bit-flip steps on the upper triangle
// ---------------------------------------------------------------------------
__global__ void qsample_kernel(const float* __restrict__ adj,
                               const int* __restrict__ tptr,
                               const uint32_t* __restrict__ bits,
                               float* __restrict__ out) {
    int idx = blockIdx.x * 256 + threadIdx.x;     // 1M threads
    int i = idx >> 10, j = idx & 1023;
    if (i > j) return;
    if (i == j) { out[idx] = 0.0f; return; }
    int t = *tptr;
    const uint32_t HALF = 524288u;                // (1024*1024)/2
    uint32_t flat = (uint32_t)idx;
    uint32_t c0, c1; bool use0;
    if (flat < HALF) { c0 = flat;        c1 = flat + HALF; use0 = true;  }
    else             { c0 = flat - HALF; c1 = flat;        use0 = false; }
    float a = adj[idx];
    for (int s = 0; s <= t; s++) {
        uint32_t o0, o1;
        threefry2x32(bits[2 * s], bits[2 * s + 1], c0, c1, o0, o1);
        uint32_t b = use0 ? o0 : o1;
        float u = __uint_as_float((b >> 9) | 0x3f800000u) - 1.0f;  // [0,1)
        float beta = 0.0001f + (0.02f - 0.0001f) * ((float)s / 99.0f);
        if (u < beta) a = 1.0f - a;   // a*(1-m) + (1-a)*m
    }
    out[idx] = a;
    out[j * 1024 + i] = a;   // symmetrize (triu + transpose)
}

// ---------------------------------------------------------------------------
// Stage 2: degree rsqrt of (A + I) row sums
// ---------------------------------------------------------------------------
__global__ void rowdeg_kernel(const float* __restrict__ a, float* __restrict__ dinv) {
    int row = blockIdx.x;
    float acc = 0.0f;
    for (int c = threadIdx.x; c < 1024; c += 256) acc += a[row * 1024 + c];
    float s = block_reduce_sum_256(acc);
    if (threadIdx.x == 0) dinv[row] = rsqrtf(s + 1.0f);
}

__global__ void anorm_kernel(const float* __restrict__ a, const float* __restrict__ dinv,
                             float* __restrict__ an) {
    int idx = blockIdx.x * 256 + threadIdx.x;
    int i = idx >> 10, j = idx & 1023;
    float v = a[idx] + (i == j ? 1.0f : 0.0f);
    an[idx] = v * dinv[i] * dinv[j];
}

// ---------------------------------------------------------------------------
// Stage 3: xw1 = x(1024x11) @ w1(11x128)   (K=11 too ragged for WMMA)
// ---------------------------------------------------------------------------
__global__ void xw1_kernel(const float* __restrict__ x, const float* __restrict__ w1,
                           float* __restrict__ out) {
    int idx = blockIdx.x * 256 + threadIdx.x;     // 131072
    int i = idx >> 7, k = idx & 127;
    float acc = 0.0f;
#pragma unroll
    for (int d = 0; d < 11; d++) acc = fmaf(x[i * 11 + d], w1[d * 128 + k], acc);
    out[idx] = acc;
}

// ---------------------------------------------------------------------------
// WMMA f32 GEMM:  C(MxN) = [relu]( A(MxK) @ B(KxN) )
// One 16x16 C tile per wave32 via V_WMMA_F32_16X16X4_F32.
// A-frag (16x4, 2 VGPR/lane): lane L -> m=L&15, k=2*(L>>4)+{0,1}
// B-frag (4x16, 2 VGPR/lane): lane L -> n=L&15, k=2*(L>>4)+{0,1}
// C/D    (16x16, 8 VGPR):     lane L, vgpr v -> m=v+8*(L>>4), n=L&15
// ---------------------------------------------------------------------------
__global__ void gemm_wmma_f32(const float* __restrict__ A, int lda,
                              const float* __restrict__ B, int ldb,
                              float* __restrict__ C, int ldc,
                              int Nn, int K, int do_relu) {
    int wave = threadIdx.x >> 5;
    int lane = threadIdx.x & 31;
    int tile = blockIdx.x * 8 + wave;
    int tilesN = Nn >> 4;
    int tm = (tile / tilesN) << 4;
    int tn = (tile % tilesN) << 4;

    int mrow = lane & 15;
    int kh = (lane >> 4) << 1;

    const float* __restrict__ arow = A + (size_t)(tm + mrow) * lda + kh;
    const float* __restrict__ bcol = B + (size_t)kh * ldb + tn + mrow;

    v8f acc = {};
    for (int k = 0; k < K; k += 4) {
        v2f a, b;
        a.x = arow[k];
        a.y = arow[k + 1];
        b.x = bcol[(size_t)k * ldb];
        b.y = bcol[(size_t)(k + 1) * ldb];
        acc = __builtin_amdgcn_wmma_f32_16x16x4_f32(
            /*neg_a=*/false, a, /*neg_b=*/false, b,
            /*c_mod=*/(short)0, acc, /*reuse_a=*/false, /*reuse_b=*/false);
    }

    int crow = (lane >> 4) * 8;
#pragma unroll
    for (int v = 0; v < 8; v++) {
        float val = acc[v];
        if (do_relu) val = fmaxf(val, 0.0f);
        C[(size_t)(tm + crow + v) * ldc + tn + (lane & 15)] = val;
    }
}

// ---------------------------------------------------------------------------
// Stage 5: base[k] = time_emb[t] @ w_t + mlp1_b
// ---------------------------------------------------------------------------
__global__ void base_kernel(const float* __restrict__ time_emb, const int* __restrict__ tptr,
                            const float* __restrict__ wt, const float* __restrict__ b1,
                            float* __restrict__ basev) {
    int k = threadIdx.x;          // 128 threads
    int t = *tptr;
    const float* te = time_emb + t * 128;
    float acc = b1[k];
    for (int d = 0; d < 128; d++) acc = fmaf(te[d], wt[d * 128 + k], acc);
    basev[k] = acc;
}

// ---------------------------------------------------------------------------
// Stage 6: P[i,j] = sigmoid( relu(hwi[i]+hwj[j]+base) . w2v + b2 )
// LDS-tiled; rows padded to 132 floats to avoid 64-bank conflicts.
// ---------------------------------------------------------------------------
__global__ void pairwise_kernel(const float* __restrict__ hwi, const float* __restrict__ hwj,
                                const float* __restrict__ basev, const float* __restrict__ w2v,
                                const float* __restrict__ b2, float* __restrict__ P) {
    __shared__ float shi[16][132];
    __shared__ float shj[16][132];
    __shared__ float sb[128];
    __shared__ float sw[128];
    int i0 = blockIdx.y * 16, j0 = blockIdx.x * 16;
    int tid = threadIdx.x;
    for (int e = tid; e < 16 * 128; e += 256) {
        int r = e >> 7, c = e & 127;
        shi[r][c] = hwi[(i0 + r) * 128 + c];
        shj[r][c] = hwj[(j0 + r) * 128 + c];
    }
    if (tid < 128) { sb[tid] = basev[tid]; sw[tid] = w2v[tid]; }
    __syncthreads();
    int li = tid >> 4, lj = tid & 15;
    float acc = b2[0];
#pragma unroll 4
    for (int k = 0; k < 128; k++) {
        float t = shi[li][k] + shj[lj][k] + sb[k];
        acc = fmaf(fmaxf(t, 0.0f), sw[k], acc);
    }
    float p = 1.0f / (1.0f + expf(-acc));
    P[(size_t)(i0 + li) * 1024 + j0 + lj] = p;
}

// ---------------------------------------------------------------------------
// Stage 7: deterministic BCE reduction
// ---------------------------------------------------------------------------
__global__ void loss_kernel(const float* __restrict__ P, const float* __restrict__ adj,
                            float* __restrict__ partials) {
    int idx = blockIdx.x * 256 + threadIdx.x;     // 262144 threads, 4 elems each
    float acc = 0.0f;
    for (int e = idx; e < 1024 * 1024; e += 1024 * 256) {
        int i = e >> 10, j = e & 1023;
        float ps = 0.5f * (P[e] + P[j * 1024 + i]);
        float a = adj[e];
        acc -= a * logf(ps + 1e-12f) + (1.0f - a) * logf(1.0f - ps + 1e-12f);
    }
    float s = block_reduce_sum_256(acc);
    if (threadIdx.x == 0) partials[blockIdx.x] = s;
}

__global__ void final_kernel(const float* __restrict__ partials, float* __restrict__ out) {
    float acc = 0.0f;
    for (int e = threadIdx.x; e < 1024; e += 256) acc += partials[e];
    float s = block_reduce_sum_256(acc);
    if (threadIdx.x == 0) out[0] = s / (1024.0f * 1024.0f);
}

// ---------------------------------------------------------------------------
// Host-side orchestration (graph-capture safe: kernels only)
// ---------------------------------------------------------------------------
extern "C" void kernel_launch(void* const* d_in, const int* in_sizes, int n_in,
                              void* d_out, int out_size, void* d_ws, size_t ws_size,
                              hipStream_t stream) {
    const float* x        = (const float*)d_in[0];   // 1024 x 11
    const float* adj      = (const float*)d_in[1];   // 1024 x 1024
    const int*   tptr     = (const int*)d_in[2];     // scalar
    const float* w1       = (const float*)d_in[3];   // 11 x 128
    const float* w2       = (const float*)d_in[4];   // 128 x 128
    const float* time_emb = (const float*)d_in[5];   // 100 x 128
    const float* mlp1_w   = (const float*)d_in[6];   // 384 x 128
    const float* mlp1_b   = (const float*)d_in[7];   // 128
    const float* mlp2_w   = (const float*)d_in[8];   // 128 (x1)
    const float* mlp2_b   = (const float*)d_in[9];   // 1
    float* out = (float*)d_out;

    // workspace carve-up (floats)
    float* f     = (float*)d_ws;
    float* adjn  = f;                         // 1048576
    float* anorm = f + 1048576;               // 1048576
    float* P     = f + 2097152;               // 1048576
    float* xw1   = f + 3145728;               // 131072
    float* h1    = xw1 + 131072;              // 131072
    float* g     = h1 + 131072;               // 131072
    float* h2    = g + 131072;                // 131072
    float* hwi   = h2 + 131072;               // 131072
    float* hwj   = hwi + 131072;              // 131072
    float* basev = hwj + 131072;              // 128
    float* dinv  = basev + 128;               // 1024
    float* parts = dinv + 1024;               // 1024
    uint32_t* bits = (uint32_t*)(parts + 1024); // 256 u32

    const float* w_i = mlp1_w;                // rows [0,128)
    const float* w_j = mlp1_w + 128 * 128;    // rows [128,256)
    const float* w_t = mlp1_w + 256 * 128;    // rows [256,384)

    subkey_kernel<<<1, 128, 0, stream>>>(tptr, bits);
    qsample_kernel<<<4096, 256, 0, stream>>>(adj, tptr, bits, adjn);
    rowdeg_kernel<<<1024, 256, 0, stream>>>(adjn, dinv);
    anorm_kernel<<<4096, 256, 0, stream>>>(adjn, dinv, anorm);
    xw1_kernel<<<512, 256, 0, stream>>>(x, w1, xw1);

    // h1 = relu(anorm @ xw1)        M=1024 N=128 K=1024 -> 512 tiles / 8 waves
    gemm_wmma_f32<<<64, 256, 0, stream>>>(anorm, 1024, xw1, 128, h1, 128, 128, 1024, 1);
    // g = h1 @ w2                   M=1024 N=128 K=128
    gemm_wmma_f32<<<64, 256, 0, stream>>>(h1, 128, w2, 128, g, 128, 128, 128, 0);
    // h2 = relu(anorm @ g)
    gemm_wmma_f32<<<64, 256, 0, stream>>>(anorm, 1024, g, 128, h2, 128, 128, 1024, 1);
    // hwi = h2 @ w_i ; hwj = h2 @ w_j
    gemm_wmma_f32<<<64, 256, 0, stream>>>(h2, 128, w_i, 128, hwi, 128, 128, 128, 0);
    gemm_wmma_f32<<<64, 256, 0, stream>>>(h2, 128, w_j, 128, hwj, 128, 128, 128, 0);

    base_kernel<<<1, 128, 0, stream>>>(time_emb, tptr, w_t, mlp1_b, basev);

    dim3 pgrid(64, 64);
    pairwise_kernel<<<pgrid, 256, 0, stream>>>(hwi, hwj, basev, mlp2_w, mlp2_b, P);

    loss_kernel<<<1024, 256, 0, stream>>>(P, adj, parts);
    final_kernel<<<1, 256, 0, stream>>>(parts, out);
}